// GunGNN_36885179138438
// MI455X (gfx1250) — compile-verified
//
#include <hip/hip_runtime.h>

typedef float v2f __attribute__((ext_vector_type(2)));
typedef float v4f __attribute__((ext_vector_type(4)));
typedef float v8f __attribute__((ext_vector_type(8)));
typedef int   v4i __attribute__((ext_vector_type(4)));

// ---------------------------------------------------------------------------
// Workspace layout (d_ws):
//   [0 .. n_gun*4)            float    summed   (per-gun 4-component sums)
//   [n_gun*4 .. n_gun*5)      uint32   counts   (per-gun edge counts)
//   [n_gun*5 .. n_gun*5+48)   float    fused weights:
//        fw[ 0..31] = A = W_l @ W_fc   (4x8, row-major)
//        fw[32..39] = v = W_r @ W_fc   (8)
//        fw[40..47] = c = b_l @ W_fc + b_fc (8)
// ---------------------------------------------------------------------------

__global__ __launch_bounds__(256)
void gun_zero_kernel(unsigned* __restrict__ ws_words, int n_words) {
    int i = blockIdx.x * blockDim.x + threadIdx.x;
    if (i < n_words) ws_words[i] = 0u;
}

__global__ __launch_bounds__(64)
void gun_prep_weights_kernel(const float* __restrict__ W_l,   // [4,64]
                             const float* __restrict__ b_l,   // [64]
                             const float* __restrict__ W_r,   // [1,64]
                             const float* __restrict__ W_fc,  // [64,8]
                             const float* __restrict__ b_fc,  // [8]
                             float* __restrict__ fw) {
    const int t = threadIdx.x;
    if (t < 32) {                       // A[i][o] = sum_h W_l[i][h] * W_fc[h][o]
        const int i = t >> 3, o = t & 7;
        float acc = 0.f;
        for (int h = 0; h < 64; ++h) acc += W_l[i * 64 + h] * W_fc[h * 8 + o];
        fw[t] = acc;
    } else if (t < 40) {                // v[o] = sum_h W_r[h] * W_fc[h][o]
        const int o = t - 32;
        float acc = 0.f;
        for (int h = 0; h < 64; ++h) acc += W_r[h] * W_fc[h * 8 + o];
        fw[32 + o] = acc;
    } else if (t < 48) {                // c[o] = sum_h b_l[h] * W_fc[h][o] + b_fc[o]
        const int o = t - 40;
        float acc = b_fc[o];
        for (int h = 0; h < 64; ++h) acc += b_l[h] * W_fc[h * 8 + o];
        fw[40 + o] = acc;
    }
}

// ---------------------------------------------------------------------------
// Edge scatter: 4 edges per thread. Edge indices are streamed once -> NT loads
// keep the 128MB of indices from evicting x_enemy / summed / counts (all
// L2-resident in 192MB). Sums use native global_atomic_add_f32.
// ---------------------------------------------------------------------------
__global__ __launch_bounds__(256)
void gun_scatter_kernel(const int* __restrict__ esrc,
                        const int* __restrict__ edst,
                        const v4f* __restrict__ x_enemy,   // [n_enemy] rows of 4
                        float* __restrict__ summed,
                        unsigned* __restrict__ counts,
                        int n_edges) {
    const int t = blockIdx.x * blockDim.x + threadIdx.x;
    int e = t * 4;
    if (e + 3 < n_edges) {
        v4i s = __builtin_nontemporal_load((const v4i*)(esrc + e));
        v4i d = __builtin_nontemporal_load((const v4i*)(edst + e));
#pragma unroll
        for (int k = 0; k < 4; ++k) {
            const int si = s[k];
            const int di = d[k];
            v4f m = x_enemy[si];                       // L2-resident gather
            float* p = summed + (size_t)di * 4;
            unsafeAtomicAdd(p + 0, m[0]);
            unsafeAtomicAdd(p + 1, m[1]);
            unsafeAtomicAdd(p + 2, m[2]);
            unsafeAtomicAdd(p + 3, m[3]);
            atomicAdd(counts + di, 1u);
        }
    } else {
        for (; e < n_edges; ++e) {
            const int si = esrc[e];
            const int di = edst[e];
            v4f m = x_enemy[si];
            float* p = summed + (size_t)di * 4;
            unsafeAtomicAdd(p + 0, m[0]);
            unsafeAtomicAdd(p + 1, m[1]);
            unsafeAtomicAdd(p + 2, m[2]);
            unsafeAtomicAdd(p + 3, m[3]);
            atomicAdd(counts + di, 1u);
        }
    }
}

// ---------------------------------------------------------------------------
// Finalize: each wave32 handles 16 guns with one V_WMMA_F32_16X16X4_F32.
//   A (16x4 mean tile):  lanes 0-15 (M=lane): VGPR0=K0, VGPR1=K1
//                        lanes 16-31 (M=lane-16): VGPR0=K2, VGPR1=K3
//   B (4x16 weights, cols 8..15 zero): N = lane&15; lo half rows K0/K1,
//                                      hi half rows K2/K3.
//   C/D (16x16 f32): VGPR r -> row M = r + (lane>=16 ? 8 : 0), col N = lane&15.
// The x_gun*v + c affine term is preloaded into C so WMMA produces the final
// values and the epilogue is pure stores. Interior/tail split is wave-uniform
// so the hot path has no per-row guards; EXEC is full at the WMMA.
// ---------------------------------------------------------------------------
__global__ __launch_bounds__(256)
void gun_finalize_kernel(const float* __restrict__ summed,
                         const unsigned* __restrict__ counts,
                         const float* __restrict__ fw,
                         const float* __restrict__ x_gun,
                         float* __restrict__ out,
                         int n_gun) {
    const int lane = threadIdx.x & 31;
    const int wave = threadIdx.x >> 5;
    const int g0 = blockIdx.x * 128 + wave * 16;   // 8 waves * 16 guns
    const int m = lane & 15;
    const bool hi = lane >= 16;
    const int g = g0 + m;
    const bool interior = (g0 + 16) <= n_gun;      // uniform across the wave

    // --- per-gun inputs (A tile + x_gun), hot path unguarded -------------
    v2f a = {0.f, 0.f};
    float xg = 0.f;
    const int k0 = hi ? 2 : 0;
    if (interior) {
        const float inv = 1.0f / fmaxf((float)counts[g], 1.0f);
        a[0] = summed[(size_t)g * 4 + k0 + 0] * inv;
        a[1] = summed[(size_t)g * 4 + k0 + 1] * inv;
        xg = x_gun[g];                             // lanes 0-15 / 16-31 same g
    } else if (g < n_gun) {
        const float inv = 1.0f / fmaxf((float)counts[g], 1.0f);
        a[0] = summed[(size_t)g * 4 + k0 + 0] * inv;
        a[1] = summed[(size_t)g * 4 + k0 + 1] * inv;
        xg = x_gun[g];
    }

    // --- B tile + per-column affine params (branchless, EXEC stays full) -
    const float b0   = (m < 8) ? fw[(hi ? 2 : 0) * 8 + m] : 0.f;
    const float b1   = (m < 8) ? fw[(hi ? 3 : 1) * 8 + m] : 0.f;
    const float vcol = (m < 8) ? fw[32 + m] : 0.f;
    const float ccol = (m < 8) ? fw[40 + m] : 0.f;
    v2f b = {b0, b1};

    // --- C = x_gun[row]*v[col] + c[col], rows fetched via ds_bpermute ----
    v8f c;
#pragma unroll
    for (int r = 0; r < 8; ++r) {
        const float xr = __shfl(xg, (hi ? 8 : 0) + r, 32);
        c[r] = xr * vcol + ccol;
    }

    c = __builtin_amdgcn_wmma_f32_16x16x4_f32(
            /*neg_a=*/false, a, /*neg_b=*/false, b,
            /*c_mod=*/(short)0, c, /*reuse_a=*/false, /*reuse_b=*/false);

    // --- epilogue: pure stores, immediate offsets off one base -----------
    const int row0 = g0 + (hi ? 8 : 0);
    if (interior) {
        if (m < 8) {
            float* p = out + (size_t)row0 * 8 + m;
#pragma unroll
            for (int r = 0; r < 8; ++r) p[r * 8] = c[r];
        }
    } else {
        if (m < 8) {
#pragma unroll
            for (int r = 0; r < 8; ++r) {
                const int gr = row0 + r;
                if (gr < n_gun) out[(size_t)gr * 8 + m] = c[r];
            }
        }
    }
}

// ---------------------------------------------------------------------------
extern "C" void kernel_launch(void* const* d_in, const int* in_sizes, int n_in,
                              void* d_out, int out_size, void* d_ws, size_t ws_size,
                              hipStream_t stream) {
    const float* x_enemy = (const float*)d_in[0];   // [n_enemy, 4]
    const float* x_gun   = (const float*)d_in[1];   // [n_gun, 1]
    const int*   esrc    = (const int*)d_in[2];     // [n_edges]
    const int*   edst    = (const int*)d_in[3];     // [n_edges]
    const float* W_l     = (const float*)d_in[4];   // [4, 64]
    const float* b_l     = (const float*)d_in[5];   // [64]
    const float* W_r     = (const float*)d_in[6];   // [1, 64]
    const float* W_fc    = (const float*)d_in[7];   // [64, 8]
    const float* b_fc    = (const float*)d_in[8];   // [8]

    const int n_gun   = in_sizes[1];
    const int n_edges = in_sizes[2];

    float*    summed = (float*)d_ws;
    unsigned* counts = (unsigned*)(summed + (size_t)n_gun * 4);
    float*    fw     = (float*)(counts + n_gun);

    // 1) zero accumulators (summed + counts = n_gun*5 words)
    {
        const int n_words = n_gun * 5;
        const int grid = (n_words + 255) / 256;
        gun_zero_kernel<<<grid, 256, 0, stream>>>((unsigned*)d_ws, n_words);
    }

    // 2) fold the linear layers into a single 4->8 affine map
    gun_prep_weights_kernel<<<1, 64, 0, stream>>>(W_l, b_l, W_r, W_fc, b_fc, fw);

    // 3) edge scatter (4 edges/thread)
    {
        const int n_thr = (n_edges + 3) / 4;
        const int grid = (n_thr + 255) / 256;
        gun_scatter_kernel<<<grid, 256, 0, stream>>>(
            esrc, edst, (const v4f*)x_enemy, summed, counts, n_edges);
    }

    // 4) WMMA finalize: 128 guns per 256-thread block (16 per wave32)
    {
        const int grid = (n_gun + 127) / 128;
        gun_finalize_kernel<<<grid, 256, 0, stream>>>(
            summed, counts, fw, x_gun, (float*)d_out, n_gun);
    }
}